// Classification_10651518894899
// MI455X (gfx1250) — compile-verified
//
#include <hip/hip_runtime.h>
#include <hip/hip_bf16.h>

typedef __attribute__((ext_vector_type(2))) float v2f;
typedef __attribute__((ext_vector_type(4))) float v4f;
typedef __attribute__((ext_vector_type(8))) float v8f;

#define SEQ_N   20
#define FRAME_F 2764800              // 3*720*1280
#define NSTEPS  (FRAME_F / 8)        // 345600 iterations of 8 columns each
#define WS_CROSS 0                   // 400 floats: cross sums
#define WS_SQ1   400                 // 20 floats: sum c1[j]^2
#define WS_SQ2   420                 // 20 floats: sum c2[i]^2
#define WS_TOTAL 440

// ---------------------------------------------------------------------------
// Zero the workspace accumulators (deterministic; runs every launch).
// ---------------------------------------------------------------------------
__global__ void zero_ws_kernel(float* __restrict__ ws) {
    int i = blockIdx.x * blockDim.x + threadIdx.x;
    if (i < WS_TOTAL) ws[i] = 0.0f;
}

// ---------------------------------------------------------------------------
// Main pass: single sweep over both 221MB clips (memory-bound, ~19us floor).
//   cross[i,j] += sum_f c2[i,f]*c1[j,f]   via v_wmma_f32_16x16x4_f32
//   sq1[j]     += sum_f c1[j,f]^2         (folded into the same loads)
//   sq2[i]     += sum_f c2[i,f]^2
//
// 20 rows covered by overlapping 16-row tiles {0..15} and {4..19}: all loads
// valid, duplicated columns hit L0, HBM traffic stays at the 442MB minimum.
//
// Each lane loads a float4 per operand per 8-column step. Since the K-sum is
// order-free, (x,y) across lanes forms a legal 16x4 fragment over columns
// {8t,8t+1,8t+4,8t+5} and (z,w) over {8t+2,8t+3,8t+6,8t+7}; A and B use the
// identical column mapping so each WMMA multiplies matching K slices.
// ---------------------------------------------------------------------------
__global__ void __launch_bounds__(256)
cross_wmma_kernel(const float* __restrict__ c1,   // clip1 [20, F]
                  const float* __restrict__ c2,   // clip2 [20, F]
                  float* __restrict__ ws) {
    __shared__ float s_cross[SEQ_N * SEQ_N];
    __shared__ float s_sq1[SEQ_N];
    __shared__ float s_sq2[SEQ_N];

    const int tid = threadIdx.x;
    for (int i = tid; i < WS_TOTAL; i += blockDim.x) {
        if (i < 400)      s_cross[i] = 0.0f;
        else if (i < 420) s_sq1[i - 400] = 0.0f;
        else              s_sq2[i - 420] = 0.0f;
    }
    __syncthreads();

    const int lane    = tid & 31;
    const int p       = lane & 15;   // row/col within tile
    const int khi     = lane >> 4;   // which K-half this lane holds
    const int wavesPerBlock = blockDim.x >> 5;
    const int gwave   = blockIdx.x * wavesPerBlock + (tid >> 5);
    const int nwaves  = gridDim.x * wavesPerBlock;

    // contiguous span of 8-column steps per wave
    const int spw = (NSTEPS + nwaves - 1) / nwaves;
    int s0 = gwave * spw;
    int s1 = s0 + spw;
    if (s0 > NSTEPS) s0 = NSTEPS;
    if (s1 > NSTEPS) s1 = NSTEPS;

    const float* a0p = c2 + (size_t)p       * FRAME_F + (size_t)(khi * 4);  // rows 0..15
    const float* a1p = c2 + (size_t)(p + 4) * FRAME_F + (size_t)(khi * 4);  // rows 4..19
    const float* b0p = c1 + (size_t)p       * FRAME_F + (size_t)(khi * 4);
    const float* b1p = c1 + (size_t)(p + 4) * FRAME_F + (size_t)(khi * 4);

    v8f acc00 = {}, acc01 = {}, acc10 = {}, acc11 = {};
    float sq1lo = 0.0f, sq1hi = 0.0f, sq2lo = 0.0f, sq2hi = 0.0f;

    #pragma unroll 2
    for (int s = s0; s < s1; ++s) {
        const size_t col = (size_t)s * 8;
        v4f A0 = *(const v4f*)(a0p + col);
        v4f A1 = *(const v4f*)(a1p + col);
        v4f B0 = *(const v4f*)(b0p + col);
        v4f B1 = *(const v4f*)(b1p + col);

        v2f a0lo = __builtin_shufflevector(A0, A0, 0, 1);
        v2f a0hi = __builtin_shufflevector(A0, A0, 2, 3);
        v2f a1lo = __builtin_shufflevector(A1, A1, 0, 1);
        v2f a1hi = __builtin_shufflevector(A1, A1, 2, 3);
        v2f b0lo = __builtin_shufflevector(B0, B0, 0, 1);
        v2f b0hi = __builtin_shufflevector(B0, B0, 2, 3);
        v2f b1lo = __builtin_shufflevector(B1, B1, 0, 1);
        v2f b1hi = __builtin_shufflevector(B1, B1, 2, 3);

        // interleave independent accumulator chains: no back-to-back RAW on D
        acc00 = __builtin_amdgcn_wmma_f32_16x16x4_f32(false, a0lo, false, b0lo, (short)0, acc00, false, false);
        acc01 = __builtin_amdgcn_wmma_f32_16x16x4_f32(false, a0lo, false, b1lo, (short)0, acc01, false, false);
        acc10 = __builtin_amdgcn_wmma_f32_16x16x4_f32(false, a1lo, false, b0lo, (short)0, acc10, false, false);
        acc11 = __builtin_amdgcn_wmma_f32_16x16x4_f32(false, a1lo, false, b1lo, (short)0, acc11, false, false);
        acc00 = __builtin_amdgcn_wmma_f32_16x16x4_f32(false, a0hi, false, b0hi, (short)0, acc00, false, false);
        acc01 = __builtin_amdgcn_wmma_f32_16x16x4_f32(false, a0hi, false, b1hi, (short)0, acc01, false, false);
        acc10 = __builtin_amdgcn_wmma_f32_16x16x4_f32(false, a1hi, false, b0hi, (short)0, acc10, false, false);
        acc11 = __builtin_amdgcn_wmma_f32_16x16x4_f32(false, a1hi, false, b1hi, (short)0, acc11, false, false);

        // row sums-of-squares. Rows 0..15 from tile0 loads; rows 16..19 come
        // from the *hi accumulators which only lanes p>=12 publish, so no
        // masking is needed during accumulation.
        sq2lo = fmaf(A0.x, A0.x, sq2lo);
        sq2lo = fmaf(A0.y, A0.y, sq2lo);
        sq2lo = fmaf(A0.z, A0.z, sq2lo);
        sq2lo = fmaf(A0.w, A0.w, sq2lo);
        sq1lo = fmaf(B0.x, B0.x, sq1lo);
        sq1lo = fmaf(B0.y, B0.y, sq1lo);
        sq1lo = fmaf(B0.z, B0.z, sq1lo);
        sq1lo = fmaf(B0.w, B0.w, sq1lo);
        sq2hi = fmaf(A1.x, A1.x, sq2hi);
        sq2hi = fmaf(A1.y, A1.y, sq2hi);
        sq2hi = fmaf(A1.z, A1.z, sq2hi);
        sq2hi = fmaf(A1.w, A1.w, sq2hi);
        sq1hi = fmaf(B1.x, B1.x, sq1hi);
        sq1hi = fmaf(B1.y, B1.y, sq1hi);
        sq1hi = fmaf(B1.z, B1.z, sq1hi);
        sq1hi = fmaf(B1.w, B1.w, sq1hi);
    }

    // ---- epilogue: scatter D tiles into shared 20x20 accumulator ----
    // C/D layout: VGPR v of lane (khi,p) holds D[M = v + 8*khi, N = p]
    // tile(0,0): rows M (0..15),  cols N (0..15)   -> all elements
    #pragma unroll
    for (int v = 0; v < 8; ++v) {
        const int M = v + 8 * khi;
        atomicAdd(&s_cross[M * SEQ_N + p], acc00[v]);
    }
    // tile(0,1): rows M, cols 4+N  -> keep cols >= 16  (p >= 12)
    if (p >= 12) {
        #pragma unroll
        for (int v = 0; v < 8; ++v) {
            const int M = v + 8 * khi;
            atomicAdd(&s_cross[M * SEQ_N + (p + 4)], acc01[v]);
        }
    }
    // tile(1,0): rows 4+M, cols N  -> keep rows >= 16  (khi==1 && v>=4)
    if (khi == 1) {
        #pragma unroll
        for (int v = 4; v < 8; ++v) {
            atomicAdd(&s_cross[(v + 12) * SEQ_N + p], acc10[v]);
        }
        // tile(1,1): rows 4+M, cols 4+N -> rows>=16 && cols>=16
        if (p >= 12) {
            #pragma unroll
            for (int v = 4; v < 8; ++v) {
                atomicAdd(&s_cross[(v + 12) * SEQ_N + (p + 4)], acc11[v]);
            }
        }
    }
    // sums of squares: lanes p and p+16 both hold partials for row p;
    // hi partials are published only by lanes p>=12 (rows 16..19).
    atomicAdd(&s_sq2[p], sq2lo);
    atomicAdd(&s_sq1[p], sq1lo);
    if (p >= 12) {
        atomicAdd(&s_sq2[p + 4], sq2hi);
        atomicAdd(&s_sq1[p + 4], sq1hi);
    }
    __syncthreads();

    // block -> global reduction
    for (int i = tid; i < 400; i += blockDim.x)
        atomicAdd(&ws[WS_CROSS + i], s_cross[i]);
    for (int i = tid; i < SEQ_N; i += blockDim.x) {
        atomicAdd(&ws[WS_SQ1 + i], s_sq1[i]);
        atomicAdd(&ws[WS_SQ2 + i], s_sq2[i]);
    }
}

// ---------------------------------------------------------------------------
// Finalize: M[i,j] = -(m2[i] + m1[j] - 2*cross[i,j]) * 1e13,
// then the 21 diagonal means for offsets -10..10.
// ---------------------------------------------------------------------------
__global__ void finalize_kernel(const float* __restrict__ ws,
                                float* __restrict__ out) {
    const int t = threadIdx.x;
    if (t > SEQ_N) return;                 // 21 outputs
    const int k  = t - (SEQ_N / 2);        // diagonal offset
    const int ak = (k < 0) ? -k : k;
    const int L  = SEQ_N - ak;
    const int r0 = (k < 0) ? -k : 0;
    const int c0 = (k > 0) ?  k : 0;
    const float invF = 1.0f / (float)FRAME_F;

    float s = 0.0f;
    for (int i = 0; i < L; ++i) {
        const int r = r0 + i;
        const int c = c0 + i;
        const float cross = ws[WS_CROSS + r * SEQ_N + c] * invF;
        const float m1    = ws[WS_SQ1 + c] * invF;
        const float m2    = ws[WS_SQ2 + r] * invF;
        s += -(m2 + m1 - 2.0f * cross) * 1.0e13f;
    }
    out[t] = s / (float)L;
}

// ---------------------------------------------------------------------------
extern "C" void kernel_launch(void* const* d_in, const int* in_sizes, int n_in,
                              void* d_out, int out_size, void* d_ws, size_t ws_size,
                              hipStream_t stream) {
    const float* clip1 = (const float*)d_in[0];
    const float* clip2 = (const float*)d_in[1];
    float* out = (float*)d_out;
    float* ws  = (float*)d_ws;

    zero_ws_kernel<<<1, 512, 0, stream>>>(ws);
    // 512 blocks x 8 waves = 4096 waves; ~85 steps (680 columns) per wave
    cross_wmma_kernel<<<512, 256, 0, stream>>>(clip1, clip2, ws);
    finalize_kernel<<<1, 32, 0, stream>>>(ws, out);
}